// NuclearLoss_30829275251212
// MI455X (gfx1250) — compile-verified
//
#include <hip/hip_runtime.h>

typedef __attribute__((ext_vector_type(2))) float v2f;
typedef __attribute__((ext_vector_type(8))) float v8f;

#define NCH 19
#define PLANE 262144              // 512*512
#define LDS_STRIDE 33             // 32 channels + 1 pad -> conflict-free

// One block = one 16x16 patch (256 threads = 8 wave32).
__global__ __launch_bounds__(256) void nuclear_patch_kernel(const float* __restrict__ x,
                                                            float* __restrict__ patch_out) {
    __shared__ float sq[256 * LDS_STRIDE];   // per-pixel p^2, 32 channels (19 real + zeros)
    __shared__ float part[8 * 32];           // per-wave per-channel partial sums
    __shared__ float red[32];                // sqrt(diag) per channel

    const int b  = blockIdx.x;               // 0..8191
    const int n  = b >> 10;                  // image
    const int pi = b & 1023;                 // patch within image
    const int ph = pi >> 5;
    const int pw = pi & 31;
    const int t  = threadIdx.x;              // 0..255 = pixel in patch
    const int h  = ph * 16 + (t >> 4);
    const int w  = pw * 16 + (t & 15);
    const float* base = x + (size_t)n * NCH * PLANE + (size_t)h * 512 + (size_t)w;

    // ---- per-pixel softmax over 19 channels (coalesced plane-strided loads) ----
    float v[NCH];
    float m = -3.4e38f;
#pragma unroll
    for (int c = 0; c < NCH; ++c) {
        v[c] = base[(size_t)c * PLANE];
        m = fmaxf(m, v[c]);
    }
    float s = 0.f;
#pragma unroll
    for (int c = 0; c < NCH; ++c) { v[c] = __expf(v[c] - m); s += v[c]; }
    const float inv = 1.0f / s;

    float* row = &sq[t * LDS_STRIDE];
#pragma unroll
    for (int c = 0; c < NCH; ++c) { float p = v[c] * inv; row[c] = p * p; }
#pragma unroll
    for (int c = NCH; c < 32; ++c) row[c] = 0.f;
    __syncthreads();

    // ---- WMMA reduction: D = ones(16x4) x B(4x16) accumulated over this wave's 32 pixels.
    // A = all ones => K-slot assignment inside B is irrelevant; only N = lane%16 matters.
    const int wave = t >> 5;
    const int lane = t & 31;
    v2f a; a.x = 1.0f; a.y = 1.0f;

#pragma unroll
    for (int cg = 0; cg < 2; ++cg) {         // channel groups 0-15, 16-31
        v8f acc = {};
        const int chan = cg * 16 + (lane & 15);
        const int poff = (lane < 16) ? 0 : 1;
#pragma unroll
        for (int j = 0; j < 8; ++j) {        // 4 pixels per WMMA, 32 per wave
            const int p0 = wave * 32 + j * 4 + poff;
            v2f bv;
            bv.x = sq[p0 * LDS_STRIDE + chan];
            bv.y = sq[(p0 + 2) * LDS_STRIDE + chan];
            acc = __builtin_amdgcn_wmma_f32_16x16x4_f32(
                false, a, false, bv, (short)0, acc, false, false);
        }
        // Diagonal of 16x16 f32 C/D: n<8 -> lane n, vgpr n ; n>=8 -> lane n+16, vgpr n-8.
        const bool act = (lane < 8) || (lane >= 24);
        const int  idx = (lane < 8) ? lane : (lane - 24);
        const int  ch  = cg * 16 + ((lane < 8) ? lane : (lane - 16));
        float d = 0.f;
#pragma unroll
        for (int i = 0; i < 8; ++i) d = (idx == i) ? acc[i] : d;
        if (act) part[wave * 32 + ch] = d;
    }
    __syncthreads();

    // ---- combine 8 waves in fixed order (deterministic), sqrt, sum over channels ----
    if (t < NCH) {
        float sum = 0.f;
#pragma unroll
        for (int wv = 0; wv < 8; ++wv) sum += part[wv * 32 + t];
        red[t] = sqrtf(sum);
    }
    __syncthreads();
    if (t == 0) {
        float total = 0.f;
#pragma unroll
        for (int c = 0; c < NCH; ++c) total += red[c];
        patch_out[b] = total;
    }
}

// Deterministic final reduction of 8192 per-patch partials.
__global__ __launch_bounds__(256) void nuclear_reduce_kernel(const float* __restrict__ pp,
                                                             float* __restrict__ out) {
    __shared__ float s[256];
    const int t = threadIdx.x;
    float acc = 0.f;
    for (int i = t; i < 8192; i += 256) acc += pp[i];
    s[t] = acc;
    __syncthreads();
#pragma unroll
    for (int k = 128; k > 0; k >>= 1) {
        if (t < k) s[t] += s[t + k];
        __syncthreads();
    }
    if (t == 0) out[0] = -s[0] * (1.0f / 8192.0f);
}

extern "C" void kernel_launch(void* const* d_in, const int* in_sizes, int n_in,
                              void* d_out, int out_size, void* d_ws, size_t ws_size,
                              hipStream_t stream) {
    const float* x   = (const float*)d_in[0];   // (8,19,512,512) f32
    float*       out = (float*)d_out;           // scalar loss
    float*       pp  = (float*)d_ws;            // 8192 per-patch partials (32 KB)

    nuclear_patch_kernel<<<8192, 256, 0, stream>>>(x, pp);
    nuclear_reduce_kernel<<<1, 256, 0, stream>>>(pp, out);
}